// Attention_12704513261859
// MI455X (gfx1250) — compile-verified
//
#include <hip/hip_runtime.h>
#include <math.h>

#define B_ 32
#define T_ 1024
#define D_ 512
#define H_ 10

typedef __attribute__((ext_vector_type(2))) float v2f;
typedef __attribute__((ext_vector_type(8))) float v8f;

// ---------------------------------------------------------------------------
// Kernel 0: r[b,h] = b1[h] + sum_d rnn[b,d] * W1[D+d, h]   (h<10, else 0)
// ---------------------------------------------------------------------------
__global__ void k_rproj(const float* __restrict__ rnn, const float* __restrict__ W1,
                        const float* __restrict__ b1, float* __restrict__ ws_r) {
    __shared__ float part[16][16];
    const int b   = blockIdx.x;
    const int tid = threadIdx.x;      // 256 threads
    const int h   = tid & 15;
    const int seg = tid >> 4;         // 16 segments of 32 d-values
    float acc = 0.f;
    const float* r = rnn + b * D_;
    if (h < H_) {
        #pragma unroll 4
        for (int i = 0; i < 32; ++i) {
            int d = seg * 32 + i;
            acc += r[d] * W1[(D_ + d) * H_ + h];
        }
    }
    part[seg][h] = acc;
    __syncthreads();
    if (tid < 16) {
        float s = 0.f;
        #pragma unroll
        for (int i = 0; i < 16; ++i) s += part[i][tid];
        ws_r[b * 16 + tid] = (tid < H_) ? (s + b1[tid]) : 0.f;
    }
}

// ---------------------------------------------------------------------------
// Kernel 1: energies[b,t] = relu( sum_h tanh( cbhg[b,t,:]@W1[:D,h] + r[b,h] )
//                                 * W2[h] + b2 )
// One wave per 16-row T-tile. WMMA f32 16x16x4: M = H(pad16), N = 16 t-rows.
//   A lane layout: lane L -> m = L%16 (=h), K pair 2*(L/16)+{0,1}
//   B lane layout: lane L -> n = L%16 (=t), K pair 2*(L/16)+{0,1}
//   C/D layout:    VGPR j, lane L -> D[m = j+8*(L/16), n = L%16]
// ---------------------------------------------------------------------------
__global__ void k_energy(const float* __restrict__ cbhg, const float* __restrict__ W1,
                         const float* __restrict__ W2, const float* __restrict__ b2,
                         const float* __restrict__ ws_r, float* __restrict__ ws_e) {
    const int wave = threadIdx.x >> 5;
    const int lane = threadIdx.x & 31;
    const int g    = blockIdx.x * 4 + wave;   // global tile id
    const int b    = g >> 6;                  // 64 tiles of 16 per batch
    const int t0   = (g & 63) << 4;
    const int half = lane >> 4;               // 0 or 1
    const int lm   = lane & 15;

    v8f c = {};
    const float* bp  = cbhg + ((long)b * T_ + (t0 + lm)) * D_ + 2 * half;
    const float* ap0 = W1 + lm;               // column h = lm of W1[:D]
    const bool hvalid = (lm < H_);

    for (int k0 = 0; k0 < D_; k0 += 4) {
        // A fragment (always in-bounds of the W1 allocation; zero-select pads
        // h>=10 without EXEC divergence around the WMMA)
        float a0 = ap0[(k0 + 2 * half) * H_];
        float a1 = ap0[(k0 + 2 * half + 1) * H_];
        v2f a; a.x = hvalid ? a0 : 0.f; a.y = hvalid ? a1 : 0.f;
        // B fragment: one float2 of this lane's cbhg row
        v2f bb; bb.x = bp[k0]; bb.y = bp[k0 + 1];
        // (neg_a, A, neg_b, B, c_mod, C, reuse_a, reuse_b)
        c = __builtin_amdgcn_wmma_f32_16x16x4_f32(false, a, false, bb,
                                                  (short)0, c, false, false);
    }

    // Fused epilogue: tanh, x W2, reduce over h (8 regs + cross-half shuffle)
    float s = 0.f;
    const float* rr = ws_r + b * 16 + 8 * half;
    #pragma unroll
    for (int j = 0; j < 8; ++j) {
        int h = j + 8 * half;
        float w2 = (h < H_) ? W2[h] : 0.f;
        s += tanhf(c[j] + rr[j]) * w2;
    }
    s += __shfl_xor(s, 16, 32);               // combine h 0-7 with h 8-15
    float e = fmaxf(s + b2[0], 0.f);          // + b2, relu
    if (lane < 16) ws_e[b * T_ + t0 + lm] = e;
}

// ---------------------------------------------------------------------------
// Kernel 2: in-place stable softmax over T per batch
// ---------------------------------------------------------------------------
__global__ void k_softmax(float* __restrict__ ws_e) {
    __shared__ float red[256];
    const int b = blockIdx.x, tid = threadIdx.x;
    float* e = ws_e + b * T_;
    float v[4];
    float m = -1e30f;
    #pragma unroll
    for (int k = 0; k < 4; ++k) { v[k] = e[tid + k * 256]; m = fmaxf(m, v[k]); }
    red[tid] = m; __syncthreads();
    for (int s = 128; s > 0; s >>= 1) {
        if (tid < s) red[tid] = fmaxf(red[tid], red[tid + s]);
        __syncthreads();
    }
    m = red[0]; __syncthreads();
    float sum = 0.f;
    #pragma unroll
    for (int k = 0; k < 4; ++k) { v[k] = __expf(v[k] - m); sum += v[k]; }
    red[tid] = sum; __syncthreads();
    for (int s = 128; s > 0; s >>= 1) {
        if (tid < s) red[tid] += red[tid + s];
        __syncthreads();
    }
    const float inv = 1.0f / red[0];
    #pragma unroll
    for (int k = 0; k < 4; ++k) e[tid + k * 256] = v[k] * inv;
}

// ---------------------------------------------------------------------------
// Kernel 3: context[b,d] = sum_t w[b,t]*cbhg[b,t,d];  plus rnn copy output
// ---------------------------------------------------------------------------
__global__ void k_context(const float* __restrict__ cbhg, const float* __restrict__ rnn,
                          const float* __restrict__ ws_w, float* __restrict__ out) {
    const int b = blockIdx.x >> 1;
    const int d = ((blockIdx.x & 1) << 8) + threadIdx.x;
    const float* w = ws_w + b * T_;
    const float* x = cbhg + (long)b * T_ * D_ + d;
    float acc = 0.f;
    #pragma unroll 8
    for (int t = 0; t < T_; ++t) acc += w[t] * x[(long)t * D_];
    out[b * D_ + d] = acc;                       // context [B,1,D]
    out[B_ * D_ + b * D_ + d] = rnn[b * D_ + d]; // rnn_reshaped [B,1,D]
}

// ---------------------------------------------------------------------------
extern "C" void kernel_launch(void* const* d_in, const int* in_sizes, int n_in,
                              void* d_out, int out_size, void* d_ws, size_t ws_size,
                              hipStream_t stream) {
    (void)in_sizes; (void)n_in; (void)out_size; (void)ws_size;
    const float* cbhg = (const float*)d_in[0];  // [B,T,D]
    const float* rnn  = (const float*)d_in[1];  // [B,D]
    const float* W1   = (const float*)d_in[2];  // [2D,H]
    const float* b1   = (const float*)d_in[3];  // [H]
    const float* W2   = (const float*)d_in[4];  // [H,1]
    const float* b2   = (const float*)d_in[5];  // [1]
    float* out  = (float*)d_out;                // [B*D context | B*D rnn]
    float* ws_r = (float*)d_ws;                 // B*16 floats
    float* ws_e = ws_r + B_ * 16;               // B*T floats (energies->weights)

    k_rproj  <<<B_,                 256, 0, stream>>>(rnn, W1, b1, ws_r);
    k_energy <<<(B_ * (T_/16)) / 4, 128, 0, stream>>>(cbhg, W1, W2, b2, ws_r, ws_e);
    k_softmax<<<B_,                 256, 0, stream>>>(ws_e);
    k_context<<<B_ * 2,             256, 0, stream>>>(cbhg, rnn, ws_e, out);
}